// gNet_78915729097077
// MI455X (gfx1250) — compile-verified
//
#include <hip/hip_runtime.h>
#include <cstdint>
#include <math.h>

// ---------------------------------------------------------------------------
// CDNA5 fp32 WMMA: D[16,16] = A[16,4] x B[4,16] + C   (wave32)
// A frag (2 VGPR): lane l: m = l&15; lanes 0-15 hold K={0,1}, lanes16-31 K={2,3}
// B frag (2 VGPR): lane l: n = l&15; lanes 0-15 hold K={0,1}, lanes16-31 K={2,3}
// C/D frag (8 VGPR): vgpr r, lane l: M = r + 8*(l>>4), N = l&15
//
// LDS tiles use a K-pair interleaved layout so each B fragment is ONE aligned
// ds_load_b64:  element (k, n) of a [K,64] tile lives at
//     lds[(k>>1)*128 + 2*n + (k&1)]
// ---------------------------------------------------------------------------
typedef __attribute__((ext_vector_type(2))) float v2f;
typedef __attribute__((ext_vector_type(8))) float v8f;

__device__ __forceinline__ v8f wmma4(v2f a, v2f b, v8f c) {
  // 8 args: (neg_a, A, neg_b, B, c_mod, C, reuse_a, reuse_b)
  return __builtin_amdgcn_wmma_f32_16x16x4_f32(false, a, false, b, (short)0, c,
                                               false, false);
}

__device__ __forceinline__ int ilv(int k, int n) {
  return ((k >> 1) << 7) + (n << 1) + (k & 1);
}

// ---------------------------------------------------------------------------
// Fused opening double layer: Y[64,L] = W1[64,64] @ relu(W0[64,64] @ X[64,L])
// Block: 256 thr (8 waves), 64-column tile. wave -> (msub = w&3, nh = w>>2).
// ---------------------------------------------------------------------------
__global__ __launch_bounds__(256) void k_open_double(
    const float* __restrict__ W0, const float* __restrict__ W1,
    const float* __restrict__ X, float* __restrict__ Y, int L) {
  __shared__ __align__(16) float ldsB[64 * 64];
  __shared__ __align__(16) float ldsH[64 * 64];
  const int tid = threadIdx.x;
  const int lane = tid & 31, wave = tid >> 5;
  const int lm = lane & 15, lh = lane >> 4;
  const int msub = wave & 3, nh = wave >> 2;
  const long Ll = L;
  const long n0 = (long)blockIdx.x * 64;
  const int c0 = (2 * nh + 0) * 16 + lm;
  const int c1 = (2 * nh + 1) * 16 + lm;

  for (int idx = tid; idx < 64 * 64; idx += 256) {
    int r = idx >> 6, j = idx & 63;
    long n = n0 + j;
    ldsB[ilv(r, j)] = (n < Ll) ? X[(long)r * Ll + n] : 0.0f;
  }
  __syncthreads();

  v8f z = {};
  v8f a0 = z, a1 = z;
  for (int ks = 0; ks < 16; ++ks) {
    const float* wp = W0 + (msub * 16 + lm) * 64 + ks * 4 + lh * 2;
    v2f a; a.x = wp[0]; a.y = wp[1];
    const int pb = (ks * 2 + lh) << 7;  // pair-row base
    a0 = wmma4(a, *(const v2f*)&ldsB[pb + 2 * c0], a0);
    a1 = wmma4(a, *(const v2f*)&ldsB[pb + 2 * c1], a1);
  }
#pragma unroll
  for (int r = 0; r < 8; r += 2) {
    // m0 = msub*16 + 8*lh + r (even), pair (m0, m0+1)
    const int pb = ((msub * 16 + 8 * lh + r) >> 1) << 7;
    v2f h0; h0.x = a0[r] > 0.f ? a0[r] : 0.f;
    h0.y = a0[r + 1] > 0.f ? a0[r + 1] : 0.f;
    *(v2f*)&ldsH[pb + 2 * c0] = h0;
    v2f h1; h1.x = a1[r] > 0.f ? a1[r] : 0.f;
    h1.y = a1[r + 1] > 0.f ? a1[r + 1] : 0.f;
    *(v2f*)&ldsH[pb + 2 * c1] = h1;
  }
  __syncthreads();

  v8f o0 = z, o1 = z;
  for (int ks = 0; ks < 16; ++ks) {
    const float* wp = W1 + (msub * 16 + lm) * 64 + ks * 4 + lh * 2;
    v2f a; a.x = wp[0]; a.y = wp[1];
    const int pb = (ks * 2 + lh) << 7;
    o0 = wmma4(a, *(const v2f*)&ldsH[pb + 2 * c0], o0);
    o1 = wmma4(a, *(const v2f*)&ldsH[pb + 2 * c1], o1);
  }
#pragma unroll
  for (int r = 0; r < 8; ++r) {
    int m = msub * 16 + r + 8 * lh;
    long n = n0 + c0;
    if (n < Ll) Y[(long)m * Ll + n] = o0[r];
    n = n0 + c1;
    if (n < Ll) Y[(long)m * Ll + n] = o1[r];
  }
}

// ---------------------------------------------------------------------------
// Fused edge layer:
//   XEC[64,E] = K2[64,128] @ relu(K1[128,192] @ concat(XN[:,row], XN[:,col], XE))
// Block: 256 thr (8 waves), 64-edge tile. Interleaved LDS tile [192,64] (48KB)
// reused for the [128,64] hidden tile after a barrier.
// ---------------------------------------------------------------------------
__global__ __launch_bounds__(256) void k_edge_fused(
    const float* __restrict__ K1, const float* __restrict__ K2,
    const float* __restrict__ XN, const float* __restrict__ XE,
    const int* __restrict__ row, const int* __restrict__ col,
    float* __restrict__ XEC, int N, int E) {
  __shared__ __align__(16) float lds[192 * 64];
  __shared__ int sRow[64];
  __shared__ int sCol[64];
  const int tid = threadIdx.x;
  const int lane = tid & 31, wave = tid >> 5;
  const int lm = lane & 15, lh = lane >> 4;
  const long e0 = (long)blockIdx.x * 64;
  const long El = E;

  if (tid < 64) {
    long e = e0 + tid;
    sRow[tid] = (e < El) ? row[e] : 0;
  } else if (tid < 128) {
    long e = e0 + (tid - 64);
    sCol[tid - 64] = (e < El) ? col[e] : 0;
  }
  __syncthreads();

  // Gather/concat tile: rows 0-63 xn[row], 64-127 xn[col], 128-191 xe
  for (int idx = tid; idx < 192 * 64; idx += 256) {
    int r = idx >> 6, j = idx & 63;
    float v;
    if (r < 64) {
      v = XN[(long)r * N + sRow[j]];
    } else if (r < 128) {
      v = XN[(long)(r - 64) * N + sCol[j]];
    } else {
      long e = e0 + j;
      v = (e < El) ? XE[(long)(r - 128) * El + e] : 0.0f;
    }
    lds[ilv(r, j)] = v;
  }
  __syncthreads();

  // Stage 1: H[128,64] = relu(K1 @ tile); wave owns msub = wave, 4 n-subtiles
  v8f z = {};
  v8f acc[4];
#pragma unroll
  for (int i = 0; i < 4; ++i) acc[i] = z;
  for (int ks = 0; ks < 48; ++ks) {
    const float* wp = K1 + (wave * 16 + lm) * 192 + ks * 4 + lh * 2;
    v2f a; a.x = wp[0]; a.y = wp[1];
    const int pb = (ks * 2 + lh) << 7;
#pragma unroll
    for (int ns = 0; ns < 4; ++ns) {
      acc[ns] = wmma4(a, *(const v2f*)&lds[pb + 2 * (ns * 16 + lm)], acc[ns]);
    }
  }
  __syncthreads();  // all waves done reading tile
#pragma unroll
  for (int ns = 0; ns < 4; ++ns) {
#pragma unroll
    for (int r = 0; r < 8; r += 2) {
      const int pb = ((wave * 16 + 8 * lh + r) >> 1) << 7;
      v2f h;
      h.x = acc[ns][r] > 0.f ? acc[ns][r] : 0.f;
      h.y = acc[ns][r + 1] > 0.f ? acc[ns][r + 1] : 0.f;
      *(v2f*)&lds[pb + 2 * (ns * 16 + lm)] = h;
    }
  }
  __syncthreads();

  // Stage 2: OUT[64,64] = K2 @ H; wave -> (msub2 = w&3, nh = w>>2)
  const int msub2 = wave & 3, nh = wave >> 2;
  const int c0 = (2 * nh + 0) * 16 + lm;
  const int c1 = (2 * nh + 1) * 16 + lm;
  v8f o0 = z, o1 = z;
  for (int ks = 0; ks < 32; ++ks) {
    const float* wp = K2 + (msub2 * 16 + lm) * 128 + ks * 4 + lh * 2;
    v2f a; a.x = wp[0]; a.y = wp[1];
    const int pb = (ks * 2 + lh) << 7;
    o0 = wmma4(a, *(const v2f*)&lds[pb + 2 * c0], o0);
    o1 = wmma4(a, *(const v2f*)&lds[pb + 2 * c1], o1);
  }
#pragma unroll
  for (int r = 0; r < 8; ++r) {
    int m = msub2 * 16 + r + 8 * lh;
    long n = e0 + c0;
    if (n < El) XEC[(long)m * El + n] = o0[r];
    n = e0 + c1;
    if (n < El) XEC[(long)m * El + n] = o1[r];
  }
}

// ---------------------------------------------------------------------------
// Single GEMM: Y[64,L] = W[64,64] @ X[64,L]   (final KNout projection)
// ---------------------------------------------------------------------------
__global__ __launch_bounds__(256) void k_gemm64(
    const float* __restrict__ W, const float* __restrict__ X,
    float* __restrict__ Y, int L) {
  __shared__ __align__(16) float ldsB[64 * 64];
  const int tid = threadIdx.x;
  const int lane = tid & 31, wave = tid >> 5;
  const int lm = lane & 15, lh = lane >> 4;
  const int msub = wave & 3, nh = wave >> 2;
  const long Ll = L;
  const long n0 = (long)blockIdx.x * 64;
  const int c0 = (2 * nh + 0) * 16 + lm;
  const int c1 = (2 * nh + 1) * 16 + lm;

  for (int idx = tid; idx < 64 * 64; idx += 256) {
    int r = idx >> 6, j = idx & 63;
    long n = n0 + j;
    ldsB[ilv(r, j)] = (n < Ll) ? X[(long)r * Ll + n] : 0.0f;
  }
  __syncthreads();

  v8f z = {};
  v8f o0 = z, o1 = z;
  for (int ks = 0; ks < 16; ++ks) {
    const float* wp = W + (msub * 16 + lm) * 64 + ks * 4 + lh * 2;
    v2f a; a.x = wp[0]; a.y = wp[1];
    const int pb = (ks * 2 + lh) << 7;
    o0 = wmma4(a, *(const v2f*)&ldsB[pb + 2 * c0], o0);
    o1 = wmma4(a, *(const v2f*)&ldsB[pb + 2 * c1], o1);
  }
#pragma unroll
  for (int r = 0; r < 8; ++r) {
    int m = msub * 16 + r + 8 * lh;
    long n = n0 + c0;
    if (n < Ll) Y[(long)m * Ll + n] = o0[r];
    n = n0 + c1;
    if (n < Ll) Y[(long)m * Ll + n] = o1[r];
  }
}

// --------------------------- elementwise helpers ---------------------------
__global__ void k_zero(float* __restrict__ p, long n) {
  long i = (long)blockIdx.x * blockDim.x + threadIdx.x;
  long st = (long)gridDim.x * blockDim.x;
  for (; i < n; i += st) p[i] = 0.0f;
}

__global__ void k_count(const int* __restrict__ row, const int* __restrict__ col,
                        float* __restrict__ cr, float* __restrict__ cc, int E) {
  int e = blockIdx.x * blockDim.x + threadIdx.x;
  if (e < E) {
    atomicAdd(&cr[row[e]], 1.0f);
    atomicAdd(&cc[col[e]], 1.0f);
  }
}

__global__ void k_scatter(const float* __restrict__ XEC,
                          const int* __restrict__ row, const int* __restrict__ col,
                          float* __restrict__ SR, float* __restrict__ SC,
                          int N, int E) {
  int e = blockIdx.x * blockDim.x + threadIdx.x;
  if (e >= E) return;
  const int r = row[e], c = col[e];
  for (int h = 0; h < 64; ++h) {
    float v = XEC[(long)h * E + e];
    atomicAdd(&SR[(long)h * N + r], v);
    atomicAdd(&SC[(long)h * N + c], v);
  }
}

__global__ void k_ln_stats(const float* __restrict__ x, long n,
                           double* __restrict__ slot) {
  __shared__ double s1[256];
  __shared__ double s2[256];
  double a = 0.0, b = 0.0;
  long st = (long)gridDim.x * blockDim.x;
  for (long i = (long)blockIdx.x * blockDim.x + threadIdx.x; i < n; i += st) {
    double v = (double)x[i];
    a += v; b += v * v;
  }
  s1[threadIdx.x] = a; s2[threadIdx.x] = b;
  __syncthreads();
  for (int s = 128; s > 0; s >>= 1) {
    if ((int)threadIdx.x < s) {
      s1[threadIdx.x] += s1[threadIdx.x + s];
      s2[threadIdx.x] += s2[threadIdx.x + s];
    }
    __syncthreads();
  }
  if (threadIdx.x == 0) {
    atomicAdd(&slot[0], s1[0]);
    atomicAdd(&slot[1], s2[0]);
  }
}

__global__ void k_ln_apply(float* __restrict__ x, long n,
                           const double* __restrict__ slot) {
  double m = slot[0] / (double)n;
  double var = slot[1] / (double)n - m * m;
  float mm = (float)m;
  float rs = (float)(1.0 / sqrt(var + 1e-5));
  long st = (long)gridDim.x * blockDim.x;
  for (long i = (long)blockIdx.x * blockDim.x + threadIdx.x; i < n; i += st)
    x[i] = (x[i] - mm) * rs;
}

// xnc = 0.5*(SR/max(cr,1) + SC/max(cc,1)); also accumulate LN stats of xnc
__global__ void k_xnc_build(const float* __restrict__ SR, const float* __restrict__ SC,
                            const float* __restrict__ CR, const float* __restrict__ CC,
                            float* __restrict__ XNC, int N,
                            double* __restrict__ slot) {
  __shared__ double s1[256];
  __shared__ double s2[256];
  const long n = (long)64 * N;
  double a = 0.0, b = 0.0;
  long st = (long)gridDim.x * blockDim.x;
  for (long g = (long)blockIdx.x * blockDim.x + threadIdx.x; g < n; g += st) {
    int node = (int)(g % (long)N);
    float cr = CR[node]; cr = cr < 1.0f ? 1.0f : cr;
    float cc = CC[node]; cc = cc < 1.0f ? 1.0f : cc;
    float v = 0.5f * (SR[g] / cr + SC[g] / cc);
    XNC[g] = v;
    double dv = (double)v;
    a += dv; b += dv * dv;
  }
  s1[threadIdx.x] = a; s2[threadIdx.x] = b;
  __syncthreads();
  for (int s = 128; s > 0; s >>= 1) {
    if ((int)threadIdx.x < s) {
      s1[threadIdx.x] += s1[threadIdx.x + s];
      s2[threadIdx.x] += s2[threadIdx.x + s];
    }
    __syncthreads();
  }
  if (threadIdx.x == 0) {
    atomicAdd(&slot[0], s1[0]);
    atomicAdd(&slot[1], s2[0]);
  }
}

// xn = relu(xn + ln(xnc)) in place
__global__ void k_node_update(float* __restrict__ XN, const float* __restrict__ XNC,
                              long n, const double* __restrict__ slot) {
  double m = slot[0] / (double)n;
  double var = slot[1] / (double)n - m * m;
  float mm = (float)m;
  float rs = (float)(1.0 / sqrt(var + 1e-5));
  long st = (long)gridDim.x * blockDim.x;
  for (long i = (long)blockIdx.x * blockDim.x + threadIdx.x; i < n; i += st) {
    float v = XN[i] + (XNC[i] - mm) * rs;
    XN[i] = v > 0.f ? v : 0.f;
  }
}

// xe = relu(xe + xec) in place
__global__ void k_edge_update(float* __restrict__ XE, const float* __restrict__ XEC,
                              long n) {
  long st = (long)gridDim.x * blockDim.x;
  for (long i = (long)blockIdx.x * blockDim.x + threadIdx.x; i < n; i += st) {
    float v = XE[i] + XEC[i];
    XE[i] = v > 0.f ? v : 0.f;
  }
}

__global__ void k_copy(float* __restrict__ dst, const float* __restrict__ src, long n) {
  long st = (long)gridDim.x * blockDim.x;
  for (long i = (long)blockIdx.x * blockDim.x + threadIdx.x; i < n; i += st)
    dst[i] = src[i];
}

// ---------------------------------------------------------------------------
extern "C" void kernel_launch(void* const* d_in, const int* in_sizes, int n_in,
                              void* d_out, int out_size, void* d_ws, size_t ws_size,
                              hipStream_t stream) {
  const float* xn_in  = (const float*)d_in[0];  // [64,N]
  const float* xe_in  = (const float*)d_in[1];  // [64,E]
  const float* KNopen = (const float*)d_in[2];  // [2,64,64]
  const float* KEopen = (const float*)d_in[3];  // [2,64,64]
  const float* KNout  = (const float*)d_in[4];  // [64,64]
  const float* KE1    = (const float*)d_in[5];  // [4,128,192]
  const float* KE2    = (const float*)d_in[6];  // [4,64,128]
  const int*   row    = (const int*)d_in[7];    // [E]
  const int*   col    = (const int*)d_in[8];    // [E]

  const int N = in_sizes[0] / 64;
  const int E = in_sizes[7];
  float* out = (float*)d_out;

  // Workspace carve-up (~256 MB)
  float* p = (float*)d_ws;
  float* XN  = p; p += (size_t)64 * N;
  float* XE  = p; p += (size_t)64 * E;
  float* XEC = p; p += (size_t)64 * E;
  float* XNC = p; p += (size_t)64 * N;
  float* SR  = p; p += (size_t)64 * N;   // SR,SC contiguous
  float* SC  = p; p += (size_t)64 * N;
  float* CR  = p; p += (size_t)N;        // CR,CC contiguous
  float* CC  = p; p += (size_t)N;
  double* STATS = (double*)(((uintptr_t)p + 7) & ~(uintptr_t)7);  // 12 doubles

  const int T = 256;
  const int GS = 2048;
  const long nN = (long)64 * N, nE = (long)64 * E;
  const int nodeBlocks = (N + 63) / 64;
  const int edgeBlocks = (E + 63) / 64;

  k_zero<<<1, 64, 0, stream>>>((float*)STATS, 24);
  k_zero<<<(2 * N + T - 1) / T, T, 0, stream>>>(CR, 2L * N);
  k_count<<<(E + T - 1) / T, T, 0, stream>>>(row, col, CR, CC, E);

  // opening layers + whole-tensor layernorm
  k_open_double<<<nodeBlocks, T, 0, stream>>>(KNopen, KNopen + 4096, xn_in, XN, N);
  k_ln_stats<<<GS, T, 0, stream>>>(XN, nN, STATS + 0);
  k_ln_apply<<<GS, T, 0, stream>>>(XN, nN, STATS + 0);

  k_open_double<<<edgeBlocks, T, 0, stream>>>(KEopen, KEopen + 4096, xe_in, XE, E);
  k_ln_stats<<<GS, T, 0, stream>>>(XE, nE, STATS + 2);
  k_ln_apply<<<GS, T, 0, stream>>>(XE, nE, STATS + 2);

  for (int l = 0; l < 4; ++l) {
    k_edge_fused<<<edgeBlocks, T, 0, stream>>>(
        KE1 + (size_t)l * 128 * 192, KE2 + (size_t)l * 64 * 128,
        XN, XE, row, col, XEC, N, E);
    k_zero<<<GS, T, 0, stream>>>(SR, 2 * nN);  // SR+SC
    k_scatter<<<(E + T - 1) / T, T, 0, stream>>>(XEC, row, col, SR, SC, N, E);
    double* slot = STATS + 4 + 2 * l;
    k_xnc_build<<<GS, T, 0, stream>>>(SR, SC, CR, CC, XNC, N, slot);
    k_node_update<<<GS, T, 0, stream>>>(XN, XNC, nN, slot);
    k_edge_update<<<GS, T, 0, stream>>>(XE, XEC, nE);
  }

  // outputs: xn_out = KNout @ xn, then xe
  k_gemm64<<<nodeBlocks, T, 0, stream>>>(KNout, XN, out, N);
  k_copy<<<GS, T, 0, stream>>>(out + nN, XE, nE);
}